// Unpool_57174604644522
// MI455X (gfx1250) — compile-verified
//
#include <hip/hip_runtime.h>
#include <hip/hip_bf16.h>

#define BB 4
#define HH 128   // node feature dim (32 float4)
#define DD 16    // edge feature dim (4 float4)

// ---------------- gfx1250 async LDS path (guarded) --------------------------
#if __has_builtin(__builtin_amdgcn_global_load_async_to_lds_b128) && \
    __has_builtin(__builtin_amdgcn_global_store_async_from_lds_b128)
#define USE_ASYNC_LDS 1
typedef int v4i __attribute__((vector_size(16)));
typedef __attribute__((address_space(3))) void  lvoid_t;
typedef __attribute__((address_space(3))) v4i   lv4i_t;
#else
#define USE_ASYNC_LDS 0
#endif

#if __has_builtin(__builtin_amdgcn_s_wait_asynccnt)
#define WAIT_ASYNC0() __builtin_amdgcn_s_wait_asynccnt(0)
#else
#define WAIT_ASYNC0() asm volatile("s_wait_asynccnt 0" ::: "memory")
#endif

// ---------------------------------------------------------------------------
// 1) membership table: member[b * nNodes + pool_indices[b][i]] = 1
// ---------------------------------------------------------------------------
__global__ void k_member(const int* __restrict__ pool,
                         unsigned char* __restrict__ member,
                         int nPooled, int nNodes) {
  int i = blockIdx.x * blockDim.x + threadIdx.x;
  int b = blockIdx.y;
  if (i < nPooled) {
    int p = pool[b * nPooled + i];
    member[(size_t)b * nNodes + p] = 1;
  }
}

// ---------------------------------------------------------------------------
// 2) new_x[b, pool[b][i], :] = x[b, i, :]   (one wave32 per 512B row)
// ---------------------------------------------------------------------------
__global__ void k_scatter_x(const float4* __restrict__ x4,
                            const int* __restrict__ pool,
                            float4* __restrict__ out4,
                            int nPooled, int nNodes) {
  int gid  = blockIdx.x * blockDim.x + threadIdx.x;
  int lane = gid & 31;
  int i    = gid >> 5;                 // pooled row within batch
  int b    = blockIdx.y;
  if (i >= nPooled) return;
  int row = b * nPooled + i;
  int p   = pool[row];
  float4 v = x4[(size_t)row * (HH / 4) + lane];        // 32 float4 per row
  out4[((size_t)b * nNodes + p) * (HH / 4) + lane] = v;
}

// ---------------------------------------------------------------------------
// 3) per-block mask counts for each batch row (ballot/popc, wave32)
// ---------------------------------------------------------------------------
__global__ void k_blocksums(const int* __restrict__ src,
                            const int* __restrict__ dst,
                            const unsigned char* __restrict__ member,
                            int* __restrict__ blockSums,
                            int E, int nNodes, int numBlk) {
  __shared__ int s_sum;
  int e = blockIdx.x * 256 + threadIdx.x;
  int s = 0, d = 0;
  if (e < E) { s = src[e]; d = dst[e]; }
  for (int b = 0; b < BB; ++b) {
    bool m = false;
    if (e < E) {
      const unsigned char* mb = member + (size_t)b * nNodes;
      m = (mb[s] != 0) && (mb[d] != 0);
    }
    if (threadIdx.x == 0) s_sum = 0;
    __syncthreads();
    unsigned long long bal = __ballot(m);
    if ((threadIdx.x & 31) == 0) atomicAdd(&s_sum, __popcll(bal));
    __syncthreads();
    if (threadIdx.x == 0) blockSums[b * numBlk + (int)blockIdx.x] = s_sum;
    __syncthreads();
  }
}

// ---------------------------------------------------------------------------
// 4) exclusive scan of blockSums (one 1024-thread block per batch row)
// ---------------------------------------------------------------------------
__global__ void k_scan_blocksums(int* __restrict__ blockSums, int numBlk) {
  __shared__ int wsum[32];
  __shared__ int s_carry;
  int* arr = blockSums + (int)blockIdx.x * numBlk;
  int lane = threadIdx.x & 31, wid = threadIdx.x >> 5;
  if (threadIdx.x == 0) s_carry = 0;
  __syncthreads();
  for (int base = 0; base < numBlk; base += 1024) {
    int idx = base + (int)threadIdx.x;
    int v = (idx < numBlk) ? arr[idx] : 0;
    int inc = v;
    #pragma unroll
    for (int dlt = 1; dlt < 32; dlt <<= 1) {
      int n = __shfl_up(inc, dlt, 32);
      if (lane >= dlt) inc += n;
    }
    if (lane == 31) wsum[wid] = inc;
    __syncthreads();
    if (wid == 0) {
      int w = wsum[lane];
      #pragma unroll
      for (int dlt = 1; dlt < 32; dlt <<= 1) {
        int n = __shfl_up(w, dlt, 32);
        if (lane >= dlt) w += n;
      }
      wsum[lane] = w;
    }
    __syncthreads();
    int waveOff = (wid == 0) ? 0 : wsum[wid - 1];
    int excl    = s_carry + waveOff + (inc - v);
    int total   = wsum[31];
    __syncthreads();
    if (idx < numBlk) arr[idx] = excl;
    if (threadIdx.x == 0) s_carry += total;
    __syncthreads();
  }
}

// ---------------------------------------------------------------------------
// 5) sel[e] = b* * EP + clip(rank_{b*}(e), 0, EP-1), b* = max masked b, or -1
// ---------------------------------------------------------------------------
__global__ void k_rank_select(const int* __restrict__ src,
                              const int* __restrict__ dst,
                              const unsigned char* __restrict__ member,
                              const int* __restrict__ blockOffs,
                              int* __restrict__ sel,
                              int E, int nNodes, int numBlk, int EP) {
  __shared__ int wcnt[8];
  int e = blockIdx.x * 256 + threadIdx.x;
  int lane = threadIdx.x & 31, wid = threadIdx.x >> 5;
  int s = 0, d = 0;
  if (e < E) { s = src[e]; d = dst[e]; }
  int choice = -1;
  for (int b = 0; b < BB; ++b) {
    bool m = false;
    if (e < E) {
      const unsigned char* mb = member + (size_t)b * nNodes;
      m = (mb[s] != 0) && (mb[d] != 0);
    }
    unsigned long long bal = __ballot(m);
    if (lane == 0) wcnt[wid] = __popcll(bal);
    __syncthreads();
    int waveOff = 0;
    for (int w = 0; w < wid; ++w) waveOff += wcnt[w];
    if (m) {
      unsigned long long lt = (1ull << lane) - 1ull;
      int local = waveOff + __popcll(bal & lt);
      int rank  = blockOffs[b * numBlk + (int)blockIdx.x] + local;
      if (rank > EP - 1) rank = EP - 1;
      choice = b * EP + rank;               // ascending b => ends at max b
    }
    __syncthreads();
  }
  if (e < E) sel[e] = choice;
}

// ---------------------------------------------------------------------------
// 6) stream new_edge_attr. 4 lanes per edge (one float4 each).
//    gfx1250 path: async-gather the 64B row into LDS (no VGPR writeback),
//    s_wait_asynccnt 0, then 4 async broadcast stores straight from LDS.
//    Masked-out edges read from a zero-filled 64B pad -> branch-free.
// ---------------------------------------------------------------------------
__global__ void k_write_edges(const int* __restrict__ sel,
                              const float4* __restrict__ ea4,
                              const float4* __restrict__ zpad4,
                              float4* __restrict__ out4, int E) {
  int tid = threadIdx.x;
  int gid = blockIdx.x * blockDim.x + tid;
  int e   = gid >> 2;
  int sub = gid & 3;
  if (e >= E) return;
  if ((tid & 63) == 0) {                   // gfx1250 speculative prefetch
    int pe = e + 8192; if (pe >= E) pe = E - 1;
    __builtin_prefetch(&sel[pe], 0, 0);
  }
  int s = sel[e];
  const float4* srcp = (s < 0) ? (zpad4 + sub)
                               : (ea4 + (size_t)s * (DD / 4) + sub);
  size_t base    = (size_t)e * (DD / 4) + sub;
  size_t strideB = (size_t)E * (DD / 4);
#if USE_ASYNC_LDS
  __shared__ float4 smem[256];             // 16B per lane
  lvoid_t* lraw  = (lvoid_t*)(void*)&smem[tid];
  lv4i_t*  lchunk = (lv4i_t*)lraw;
  __builtin_amdgcn_global_load_async_to_lds_b128(
      (v4i*)(const void*)srcp, lchunk, 0, 0);
  WAIT_ASYNC0();                           // load visible in LDS
  #pragma unroll
  for (int bb = 0; bb < BB; ++bb) {
    __builtin_amdgcn_global_store_async_from_lds_b128(
        (v4i*)(void*)&out4[base + (size_t)bb * strideB], lchunk, 0, 0);
  }
  // stores drain via implicit wait-idle at s_endpgm
#else
  float4 v = *srcp;
  #pragma unroll
  for (int bb = 0; bb < BB; ++bb) out4[base + (size_t)bb * strideB] = v;
#endif
}

// ---------------------------------------------------------------------------
extern "C" void kernel_launch(void* const* d_in, const int* in_sizes, int n_in,
                              void* d_out, int out_size, void* d_ws, size_t ws_size,
                              hipStream_t stream) {
  const float* x    = (const float*)d_in[0];
  const float* ea   = (const float*)d_in[1];
  const int*   ei   = (const int*)d_in[2];   // [2, E] : src then dst
  const int*   pool = (const int*)d_in[3];   // [B, N_POOLED]
  (void)n_in; (void)ws_size;

  const int nPooled = in_sizes[3] / BB;                 // 50000
  const int E       = in_sizes[2] / 2;                  // 1600000
  const int EP      = in_sizes[1] / (BB * DD);          // 800000
  const long long edgeOutFloats = (long long)BB * E * DD;
  const int nNodes  = (int)(((long long)out_size - edgeOutFloats) /
                            ((long long)BB * HH));      // 100000

  float* newX  = (float*)d_out;
  float* newEA = (float*)d_out + (size_t)BB * nNodes * HH;

  // workspace layout: member | blockSums | sel | zero pad
  char* ws = (char*)d_ws;
  unsigned char* member = (unsigned char*)ws;
  size_t memberBytes = (((size_t)BB * nNodes) + 4095) & ~(size_t)4095;
  const int numBlk = (E + 255) / 256;
  int* blockSums = (int*)(ws + memberBytes);
  size_t bsBytes = (((size_t)BB * numBlk * sizeof(int)) + 4095) & ~(size_t)4095;
  int* sel = (int*)(ws + memberBytes + bsBytes);        // E ints
  size_t selBytes = (((size_t)E * sizeof(int)) + 4095) & ~(size_t)4095;
  float* zpad = (float*)(ws + memberBytes + bsBytes + selBytes);  // 256B zeros

  (void)hipMemsetAsync(member, 0, (size_t)BB * nNodes, stream);
  (void)hipMemsetAsync(zpad, 0, 256, stream);
  (void)hipMemsetAsync(newX, 0, (size_t)BB * nNodes * HH * sizeof(float), stream);

  dim3 gM((nPooled + 255) / 256, BB);
  k_member<<<gM, 256, 0, stream>>>(pool, member, nPooled, nNodes);

  dim3 gX((nPooled * 32 + 255) / 256, BB);              // one wave per row
  k_scatter_x<<<gX, 256, 0, stream>>>(
      (const float4*)x, pool, (float4*)newX, nPooled, nNodes);

  k_blocksums<<<numBlk, 256, 0, stream>>>(ei, ei + E, member, blockSums,
                                          E, nNodes, numBlk);
  k_scan_blocksums<<<BB, 1024, 0, stream>>>(blockSums, numBlk);
  k_rank_select<<<numBlk, 256, 0, stream>>>(ei, ei + E, member, blockSums,
                                            sel, E, nNodes, numBlk, EP);

  const long long threadsD = (long long)E * 4;
  k_write_edges<<<(int)((threadsD + 255) / 256), 256, 0, stream>>>(
      sel, (const float4*)ea, (const float4*)zpad, (float4*)newEA, E);
}